// RNNDecoder_18098992185720
// MI455X (gfx1250) — compile-verified
//
#include <hip/hip_runtime.h>
#include <cstdint>

// ---------------- types / target helpers ----------------
typedef float v2f __attribute__((ext_vector_type(2)));
typedef float v4f __attribute__((ext_vector_type(4)));
typedef float v8f __attribute__((ext_vector_type(8)));
typedef int   vs4i __attribute__((vector_size(16)));   // matches async-LDS builtin pointee

#define GLB_AS __attribute__((address_space(1)))
#define LDS_AS __attribute__((address_space(3)))

#if defined(__AMDGCN__)
  #if __has_builtin(__builtin_amdgcn_global_load_async_to_lds_b128)
    #define USE_ASYNC_LDS 1
  #endif
#endif
#ifndef USE_ASYNC_LDS
  #define USE_ASYNC_LDS 0
#endif

// CPol temporal hint: TH=NT (non-temporal) — 480MB single-pass stream > 192MB L2
#define CPOL_NT 1

#define DIM 300
#define WAVES_PER_BLOCK 2
#define TPB (WAVES_PER_BLOCK * 32)              // 64 threads
#define ROWS_PER_BLOCK (WAVES_PER_BLOCK * 16)   // 32 rows per block
#define F4_PER_BLOCK (ROWS_PER_BLOCK * DIM / 4) // 2400 float4 per tile

// ---------------- kernel 1: cosine scores via WMMA ----------------
__global__ __launch_bounds__(TPB) void cos_scores_kernel(
    const float* __restrict__ X, const float* __restrict__ w,
    float* __restrict__ scores)
{
    __shared__ float sx[ROWS_PER_BLOCK * DIM]; // 38400 B
    __shared__ float sw[DIM];                  // query vector

    const int tid  = threadIdx.x;
    const int lane = tid & 31;
    const int wave = tid >> 5;

    const long tileBase = (long)blockIdx.x * ROWS_PER_BLOCK;
    const float* src = X + tileBase * DIM;

    for (int i = tid; i < DIM; i += TPB) sw[i] = w[i];

#if USE_ASYNC_LDS
    #pragma unroll 4
    for (int i = tid; i < F4_PER_BLOCK; i += TPB) {
        __builtin_amdgcn_global_load_async_to_lds_b128(
            (GLB_AS vs4i*)(uintptr_t)(src + i * 4),
            (LDS_AS vs4i*)(uint32_t)(uintptr_t)(sx + i * 4),
            0, CPOL_NT);
    }
  #if defined(__AMDGCN__) && __has_builtin(__builtin_amdgcn_s_wait_asynccnt)
    __builtin_amdgcn_s_wait_asynccnt(0);
  #else
    asm volatile("s_wait_asynccnt 0" ::: "memory");
  #endif
#else
    for (int i = tid; i < F4_PER_BLOCK; i += TPB) {
        v4f v = __builtin_nontemporal_load(((const v4f*)src) + i);
        ((v4f*)sx)[i] = v;
    }
#endif
    __syncthreads();

    // wave-uniform ||w|| (no eps on query norm, matching reference)
    float qq = 0.f;
    for (int i = lane; i < DIM; i += 32) qq += sw[i] * sw[i];
    #pragma unroll
    for (int off = 16; off > 0; off >>= 1) qq += __shfl_xor(qq, off);
    const float qn = sqrtf(qq);

    // WMMA f32 16x16x4 A-fragment ownership (ISA 7.12.2):
    // lane L: row = L&15; lanes 0-15 hold K-pair {k,k+1}, lanes 16-31 hold {k+2,k+3}
    const int grp = lane >> 4;
    const int row = lane & 15;
    const float* a_base = sx + (wave * 16 + row) * DIM;

    v8f acc = {0.f, 0.f, 0.f, 0.f, 0.f, 0.f, 0.f, 0.f};
    float ss = 0.f;

    #pragma unroll 5
    for (int k = 0; k < DIM; k += 4) {
        const int col = k + 2 * grp;
        v2f a = *(const v2f*)(a_base + col); // ds_load_b64 (8B aligned: col even)
        v2f b = *(const v2f*)(sw + col);     // B columns identical = w chunk
        ss = fmaf(a.x, a.x, ss);
        ss = fmaf(a.y, a.y, ss);
        // D = A(16x4) * B(4x16) + C ; every column of C = partial dots
        acc = __builtin_amdgcn_wmma_f32_16x16x4_f32(
            false, a, false, b, (short)0, acc, false, false);
    }

    // full row sum-of-squares: combine the two K-pair halves
    ss += __shfl_xor(ss, 16);

    // C/D layout: lane L, vgpr r holds dot[row m = r + 8*grp] (all columns equal)
    float c[8];
    #pragma unroll
    for (int r = 0; r < 8; ++r) {
        float ssr = __shfl(ss, grp * 8 + r); // sumsq of row (grp*8 + r)
        c[r] = acc[r] / (sqrtf(ssr + 1e-9f) * qn);
    }

    // lanes 0 and 16 of each wave hold cos for rows 0..7 / 8..15 -> 2x b128 stores
    if (row == 0) {
        v4f lo = {c[0], c[1], c[2], c[3]};
        v4f hi = {c[4], c[5], c[6], c[7]};
        v4f* dst = (v4f*)(scores + tileBase + wave * 16 + grp * 8);
        dst[0] = lo;
        dst[1] = hi;
    }
}

// ---------------- kernel 2: top-10 over 400000 scores ----------------
#define TK_T 512

__global__ __launch_bounds__(TK_T) void topk_kernel(
    const float* __restrict__ s, float* __restrict__ out, int n)
{
    __shared__ float cv[TK_T * 10];
    __shared__ int   ci[TK_T * 10];
    __shared__ float rv[TK_T];
    __shared__ int   ri[TK_T];

    const int tid = threadIdx.x;
    const float NEG = -__builtin_inff();

    float lv[10];
    int   li[10];
    #pragma unroll
    for (int j = 0; j < 10; ++j) { lv[j] = NEG; li[j] = 0; }

    // register-resident sorted (desc) top-10, fully unrolled insertion
    for (int i = tid; i < n; i += TK_T) {
        float v = s[i];
        if (v > lv[9]) {
            #pragma unroll
            for (int j = 9; j >= 1; --j) {
                bool here     = v > lv[j];
                bool fromPrev = v > lv[j - 1];
                float nv = fromPrev ? lv[j - 1] : v;
                int   ni = fromPrev ? li[j - 1] : i;
                if (here) { lv[j] = nv; li[j] = ni; }
            }
            if (v > lv[0]) { lv[0] = v; li[0] = i; }
        }
    }

    #pragma unroll
    for (int j = 0; j < 10; ++j) {
        cv[tid * 10 + j] = lv[j];
        ci[tid * 10 + j] = li[j];
    }
    __syncthreads();

    // 10 rounds of block-wide argmax over the 5120 candidates
    for (int round = 0; round < 10; ++round) {
        float bm = NEG;
        int   bj = 0;
        for (int j = tid; j < TK_T * 10; j += TK_T) {
            if (cv[j] > bm) { bm = cv[j]; bj = j; }
        }
        rv[tid] = bm;
        ri[tid] = bj;
        __syncthreads();
        for (int st = TK_T / 2; st > 0; st >>= 1) {
            if (tid < st && rv[tid + st] > rv[tid]) {
                rv[tid] = rv[tid + st];
                ri[tid] = ri[tid + st];
            }
            __syncthreads();
        }
        if (tid == 0) {
            int j = ri[0];
            out[round]      = rv[0];           // value
            out[10 + round] = (float)ci[j];    // index (exact: < 2^24)
            cv[j] = NEG;                       // remove winner
        }
        __syncthreads();
    }
}

// ---------------- launch ----------------
extern "C" void kernel_launch(void* const* d_in, const int* in_sizes, int n_in,
                              void* d_out, int out_size, void* d_ws, size_t ws_size,
                              hipStream_t stream)
{
    const float* X = (const float*)d_in[0];  // [V, 300] fp32
    const float* w = (const float*)d_in[1];  // [300]    fp32
    const int V = in_sizes[0] / DIM;         // 400000

    float* scores = (float*)d_ws;            // V floats of scratch

    const int blocks = V / ROWS_PER_BLOCK;   // 12500
    cos_scores_kernel<<<blocks, TPB, 0, stream>>>(X, w, scores);
    topk_kernel<<<1, TK_T, 0, stream>>>(scores, (float*)d_out, V);
}